// MinkowskiInstanceNorm_75883482186010
// MI455X (gfx1250) — compile-verified
//
#include <hip/hip_runtime.h>

typedef __attribute__((ext_vector_type(2))) float v2f;
typedef __attribute__((ext_vector_type(4))) float v4f;
typedef __attribute__((ext_vector_type(8))) float v8f;

#define N_INST 16
#define CCH 32
#define EPSV 1e-8f

// ---------------------------------------------------------------------------
// Zero the accumulator region of the workspace (gsum 1024 + gcount 16 floats)
// ---------------------------------------------------------------------------
__global__ void zero_ws_kernel(float* ws) {
  int t = blockIdx.x * blockDim.x + threadIdx.x;
  if (t < N_INST * 64 + N_INST) ws[t] = 0.0f;
}

// ---------------------------------------------------------------------------
// Pass 1: per-instance sum / sumsq / count via WMMA ones-matmul reduction.
// Each wave owns a contiguous chunk of points (batch_ids sorted => mostly one
// instance per wave). 4-point x 32-channel tiles are reduced with
// V_WMMA_F32_16X16X4_F32 (A = all-ones), boundary tiles use a scalar path.
// Wave flushes to LDS (ds_add_f32); block flushes touched instances to global.
// ---------------------------------------------------------------------------
__global__ __launch_bounds__(256) void stats_kernel(
    const float* __restrict__ feat, const int* __restrict__ ids,
    float* __restrict__ gsum,    // [16*64]: per inst 32 sums then 32 sumsqs
    float* __restrict__ gcount,  // [16]
    int n) {
  __shared__ float lsum[N_INST * 64];
  __shared__ float lcount[N_INST];
  __shared__ int ltouched;
  for (int i = threadIdx.x; i < N_INST * 64; i += blockDim.x) lsum[i] = 0.0f;
  if (threadIdx.x < N_INST) lcount[threadIdx.x] = 0.0f;
  if (threadIdx.x == 0) ltouched = 0;
  __syncthreads();

  const int lane = threadIdx.x & 31;
  const int wave = (blockIdx.x * blockDim.x + threadIdx.x) >> 5;
  const int nwaves = (gridDim.x * blockDim.x) >> 5;
  int chunk = ((n + nwaves * 4 - 1) / (nwaves * 4)) * 4;
  int start = wave * chunk;
  int end = min(start + chunk, n);

  const int half = lane >> 4;  // 0: lanes 0-15 (K=0,1), 1: lanes 16-31 (K=2,3)
  const int nn = lane & 15;    // B-matrix column = channel within 16-ch half

  v2f ones;
  ones.x = 1.0f;
  ones.y = 1.0f;
  v8f cs_lo = {}, cs_hi = {}, cq_lo = {}, cq_hi = {};
  float cnt = 0.0f;
  int cur = -1;

  auto flush = [&]() {
    if (cur >= 0 && cnt > 0.0f) {
      if (lane < 16) {  // D[0,n] lives in VGPR0 of lanes 0-15
        atomicAdd(&lsum[cur * 64 + nn], cs_lo[0]);
        atomicAdd(&lsum[cur * 64 + 16 + nn], cs_hi[0]);
        atomicAdd(&lsum[cur * 64 + 32 + nn], cq_lo[0]);
        atomicAdd(&lsum[cur * 64 + 48 + nn], cq_hi[0]);
      }
      if (lane == 0) {
        atomicAdd(&lcount[cur], cnt);
        atomicOr(&ltouched, 1 << cur);
      }
      cs_lo = {}; cs_hi = {}; cq_lo = {}; cq_hi = {};
      cnt = 0.0f;
    }
  };

  int p = start;
  while (p < end) {
    if (p + 4 <= end) {
      int id0 = ids[p];
      int id3 = ids[p + 3];
      if (id0 == id3) {  // uniform 4-point tile (sorted ids)
        if (id0 != cur) { flush(); cur = id0; }
        // B layout (32-bit, 4x16): VGPR0 = K=0 (lanes0-15)/K=2 (lanes16-31),
        //                          VGPR1 = K=1 / K=3. N = lane%16.
        const float* r0 = feat + (size_t)(p + 2 * half) * CCH;
        const float* r1 = r0 + CCH;
        v2f blo, bhi, blo2, bhi2;
        blo.x = r0[nn];      blo.y = r1[nn];
        bhi.x = r0[16 + nn]; bhi.y = r1[16 + nn];
        blo2.x = blo.x * blo.x; blo2.y = blo.y * blo.y;
        bhi2.x = bhi.x * bhi.x; bhi2.y = bhi.y * bhi.y;
        cs_lo = __builtin_amdgcn_wmma_f32_16x16x4_f32(false, ones, false, blo,
                                                      (short)0, cs_lo, false, false);
        cs_hi = __builtin_amdgcn_wmma_f32_16x16x4_f32(false, ones, false, bhi,
                                                      (short)0, cs_hi, false, false);
        cq_lo = __builtin_amdgcn_wmma_f32_16x16x4_f32(false, ones, false, blo2,
                                                      (short)0, cq_lo, false, false);
        cq_hi = __builtin_amdgcn_wmma_f32_16x16x4_f32(false, ones, false, bhi2,
                                                      (short)0, cq_hi, false, false);
        cnt += 4.0f;
        p += 4;
        continue;
      }
    }
    // scalar path: instance boundary inside tile, or tail
    int idp = ids[p];
    if (idp != cur) { flush(); cur = idp; }
    float xlo = feat[(size_t)p * CCH + nn];
    float xhi = feat[(size_t)p * CCH + 16 + nn];
    cs_lo[0] += xlo;
    cs_hi[0] += xhi;
    cq_lo[0] += xlo * xlo;
    cq_hi[0] += xhi * xhi;
    cnt += 1.0f;
    p += 1;
  }
  flush();

  __syncthreads();
  int touched = ltouched;
  for (int i = threadIdx.x; i < N_INST * 64; i += blockDim.x) {
    if (touched & (1 << (i >> 6))) atomicAdd(&gsum[i], lsum[i]);
  }
  if (threadIdx.x < N_INST && (touched & (1 << threadIdx.x)))
    atomicAdd(&gcount[threadIdx.x], lcount[threadIdx.x]);
}

// ---------------------------------------------------------------------------
// Pass 1.5: fold stats + weight/bias into per-(instance,channel) scale/shift
// ---------------------------------------------------------------------------
__global__ void finalize_kernel(const float* __restrict__ gsum,
                                const float* __restrict__ gcount,
                                const float* __restrict__ w,
                                const float* __restrict__ b,
                                float* __restrict__ gscale,
                                float* __restrict__ gshift) {
  int t = threadIdx.x;
  if (t >= N_INST * CCH) return;
  int inst = t >> 5;
  int c = t & 31;
  float cntv = fmaxf(gcount[inst], 1.0f);
  float s = gsum[inst * 64 + c];
  float sq = gsum[inst * 64 + 32 + c];
  float mean = s / cntv;
  float var = fmaxf(sq / cntv - mean * mean, 0.0f);
  float inv = rsqrtf(var + EPSV);
  float sc = inv * w[c];
  gscale[t] = sc;
  gshift[t] = fmaf(-mean, sc, b[c]);
}

// ---------------------------------------------------------------------------
// Pass 2: streaming normalize. One float4 (b128) per thread, perfectly
// coalesced, non-temporal load/store (pure streaming traffic).
// ---------------------------------------------------------------------------
__global__ __launch_bounds__(256) void norm_kernel(
    const v4f* __restrict__ feat4, const int* __restrict__ ids,
    const float* __restrict__ gscale, const float* __restrict__ gshift,
    v4f* __restrict__ out4, int n4) {
  int gid = blockIdx.x * blockDim.x + threadIdx.x;
  if (gid >= n4) return;
  int point = gid >> 3;        // 8 float4 per 32-channel row
  int c4 = (gid & 7) * 4;
  int inst = ids[point];
  v4f x = __builtin_nontemporal_load(&feat4[gid]);
  v4f s = *(const v4f*)(gscale + inst * CCH + c4);
  v4f sh = *(const v4f*)(gshift + inst * CCH + c4);
  v4f y;
  y.x = fmaf(x.x, s.x, sh.x);
  y.y = fmaf(x.y, s.y, sh.y);
  y.z = fmaf(x.z, s.z, sh.z);
  y.w = fmaf(x.w, s.w, sh.w);
  __builtin_nontemporal_store(y, &out4[gid]);
}

extern "C" void kernel_launch(void* const* d_in, const int* in_sizes, int n_in,
                              void* d_out, int out_size, void* d_ws, size_t ws_size,
                              hipStream_t stream) {
  const float* feat = (const float*)d_in[0];
  const int* ids = (const int*)d_in[1];
  const float* weight = (const float*)d_in[2];
  const float* bias = (const float*)d_in[3];
  float* out = (float*)d_out;
  float* ws = (float*)d_ws;

  int total = in_sizes[0];  // N * 32
  int n = total / CCH;

  float* gsum = ws;             // 1024 floats
  float* gcount = ws + 1024;    // 16 floats
  float* gscale = ws + 1040;    // 512 floats (16B aligned: 4160 bytes)
  float* gshift = ws + 1552;    // 512 floats (16B aligned: 6208 bytes)

  zero_ws_kernel<<<5, 256, 0, stream>>>(ws);
  stats_kernel<<<1024, 256, 0, stream>>>(feat, ids, gsum, gcount, n);
  finalize_kernel<<<1, 512, 0, stream>>>(gsum, gcount, weight, bias, gscale, gshift);
  int n4 = total / 4;
  norm_kernel<<<(n4 + 255) / 256, 256, 0, stream>>>((const v4f*)feat, ids, gscale,
                                                    gshift, (v4f*)out, n4);
}